// GridInterpolator_39118562132123
// MI455X (gfx1250) — compile-verified
//
#include <hip/hip_runtime.h>

// Problem constants (from reference): S=16, G=48, B=16, N=16384, K=16, C=3
#define G_    48
#define N_    16384
#define KC    48                 // K*C floats per grid node (192 bytes, contiguous)
#define PTS   (16 * 16384)       // B*N = 262144 points
#define WAVES_PER_BLOCK 4
#define THREADS (WAVES_PER_BLOCK * 32)
#define LDS_STRIDE 52            // padded per-lane slice (floats): 2-way bank conflicts instead of 8-way

// Async gather: per lane, copies 16 bytes from global address gaddr + IMM to LDS offset ldsOff + IMM.
// IOFFSET applies to BOTH the LDS and global address (CDNA5 ISA),
// so one LDS-offset VGPR + one 64-bit address pair covers a whole 192B block.
#define ASYNC_LD16(ldsOff, gaddr, IMM)                                        \
  asm volatile("global_load_async_to_lds_b128 %0, %1, off offset:" #IMM      \
               :: "v"(ldsOff), "v"(gaddr) : "memory")

#define WAIT_ASYNC(N) asm volatile("s_wait_asynccnt " #N ::: "memory")
#define WAIT_DS0      asm volatile("s_wait_dscnt 0" ::: "memory")

__device__ __forceinline__ void issue_corner(unsigned ldsOff, const float* g) {
  ASYNC_LD16(ldsOff, g, 0);
  ASYNC_LD16(ldsOff, g, 16);
  ASYNC_LD16(ldsOff, g, 32);
  ASYNC_LD16(ldsOff, g, 48);
  ASYNC_LD16(ldsOff, g, 64);
  ASYNC_LD16(ldsOff, g, 80);
  ASYNC_LD16(ldsOff, g, 96);
  ASYNC_LD16(ldsOff, g, 112);
  ASYNC_LD16(ldsOff, g, 128);
  ASYNC_LD16(ldsOff, g, 144);
  ASYNC_LD16(ldsOff, g, 160);
  ASYNC_LD16(ldsOff, g, 176);
}

__global__ __launch_bounds__(THREADS) void GridInterpolator_39118562132123_kernel(
    const float* __restrict__ x,        // (B, N, 3)
    const int*   __restrict__ sArr,     // (B,)
    const float* __restrict__ values,   // (S, G, G, G, K, C)
    const float* __restrict__ center,   // (S, 3)
    const float* __restrict__ radius,   // (S,)
    const float* __restrict__ bbox,     // (S, 2, 3)
    const float* __restrict__ spacing,  // (S, 3)
    float* __restrict__ out)            // (B, N, K, C)
{
  // Double-buffered per-lane staging: 2 bufs x 4 waves x 32 lanes x 52 floats = 52KB
  __shared__ float lds[2][WAVES_PER_BLOCK][32][LDS_STRIDE];

  const int tid  = blockIdx.x * THREADS + threadIdx.x;  // point id, exact cover
  const int wave = threadIdx.x >> 5;
  const int lane = threadIdx.x & 31;

  // Raw LDS byte offsets for the async ops (flat addr low 32 bits == LDS offset).
  const unsigned ldsA = (unsigned)(size_t)&lds[0][wave][lane][0];
  const unsigned ldsB = (unsigned)(size_t)&lds[1][wave][lane][0];

  const int b = tid >> 14;          // tid / N_
  const int s = sArr[b];

  const float cx = center[s * 3 + 0];
  const float cy = center[s * 3 + 1];
  const float cz = center[s * 3 + 2];
  const float r  = radius[s];
  const float b0x = bbox[s * 6 + 0], b0y = bbox[s * 6 + 1], b0z = bbox[s * 6 + 2];
  const float spx = spacing[s * 3 + 0], spy = spacing[s * 3 + 1], spz = spacing[s * 3 + 2];

  const float px = x[tid * 3 + 0];
  const float py = x[tid * 3 + 1];
  const float pz = x[tid * 3 + 2];

  // Clamp into sphere, then map to grid index space.
  const float dx = px - cx, dy = py - cy, dz = pz - cz;
  const float nrm   = sqrtf(dx * dx + dy * dy + dz * dz);
  const float normd = fmaxf(nrm / r, 1.0f);
  const float inv   = 1.0f / normd;
  const float xc = dx * inv + cx, yc = dy * inv + cy, zc = dz * inv + cz;
  const float fx = (xc - b0x) / spx;
  const float fy = (yc - b0y) / spy;
  const float fz = (zc - b0z) / spz;
  const float flx = floorf(fx), fly = floorf(fy), flz = floorf(fz);
  const int ix = (int)flx, iy = (int)fly, iz = (int)flz;
  const float wx = fx - flx, wy = fy - fly, wz = fz - flz;
  const float nwx = 1.0f - wx, nwy = 1.0f - wy, nwz = 1.0f - wz;

  // Strides in floats
  const int strS = G_ * G_ * G_ * KC;   // 5,308,416
  const int strX = G_ * G_ * KC;        // 110,592
  const int strY = G_ * KC;             // 2,304
  const int strZ = KC;                  // 48
  const float* base = values + (size_t)s * strS + ix * strX + iy * strY + iz * strZ;

  const float* gp[8];
  float w8[8];
#pragma unroll
  for (int c = 0; c < 8; ++c) {
    const int ccx = (c >> 2) & 1, ccy = (c >> 1) & 1, ccz = c & 1;
    gp[c] = base + ccx * strX + ccy * strY + ccz * strZ;
    w8[c] = (ccx ? wx : nwx) * (ccy ? wy : nwy) * (ccz ? wz : nwz);
  }

  float4 acc[12];
#pragma unroll
  for (int j = 0; j < 12; ++j) acc[j] = make_float4(0.f, 0.f, 0.f, 0.f);

  // Software pipeline: issue corner c+1 while consuming corner c from LDS.
  issue_corner(ldsA, gp[0]);
#pragma unroll
  for (int c = 0; c < 8; ++c) {
    const int buf = c & 1;
    if (c < 7) {
      WAIT_DS0;  // prior ds_loads from the buffer we're about to overwrite are done
      issue_corner(buf ? ldsA : ldsB, gp[c + 1]);
      WAIT_ASYNC(12);  // corner c's 12 async ops complete; c+1's 12 still in flight
    } else {
      WAIT_ASYNC(0);
    }
    const float w = w8[c];
    const float4* vv = (const float4*)&lds[buf][wave][lane][0];
#pragma unroll
    for (int j = 0; j < 12; ++j) {
      const float4 v = vv[j];
      acc[j].x = fmaf(w, v.x, acc[j].x);
      acc[j].y = fmaf(w, v.y, acc[j].y);
      acc[j].z = fmaf(w, v.z, acc[j].z);
      acc[j].w = fmaf(w, v.w, acc[j].w);
    }
  }

  // 48 contiguous floats per point -> 12 x global_store_b128
  float4* op = (float4*)(out + (size_t)tid * KC);
#pragma unroll
  for (int j = 0; j < 12; ++j) op[j] = acc[j];
}

extern "C" void kernel_launch(void* const* d_in, const int* in_sizes, int n_in,
                              void* d_out, int out_size, void* d_ws, size_t ws_size,
                              hipStream_t stream) {
  const float* x       = (const float*)d_in[0];
  const int*   s       = (const int*)d_in[1];
  const float* values  = (const float*)d_in[2];
  const float* center  = (const float*)d_in[3];
  const float* radius  = (const float*)d_in[4];
  const float* bbox    = (const float*)d_in[5];
  const float* spacing = (const float*)d_in[6];
  float* out = (float*)d_out;

  const int blocks = PTS / THREADS;  // 262144 / 128 = 2048, exact
  GridInterpolator_39118562132123_kernel<<<blocks, THREADS, 0, stream>>>(
      x, s, values, center, radius, bbox, spacing, out);
}